// PieceVectorExtractor_72499047957087
// MI455X (gfx1250) — compile-verified
//
#include <hip/hip_runtime.h>

typedef __attribute__((ext_vector_type(2))) float v2f;
typedef __attribute__((ext_vector_type(8))) float v8f;

#define C_DIM 10               // input channels
#define P_DIM 32               // pieces
#define F_DIM 24               // output features
#define SQ    64               // squares per board
#define WAVES_PER_BLOCK 8      // 256 threads = 8 wave32

__global__ __launch_bounds__(256) void
piece_extract_wmma(const float* __restrict__ board,   // (N, 10, 64)
                   const int*   __restrict__ ids,     // (N, 64)
                   const float* __restrict__ Wm,      // (24, 10)
                   const float* __restrict__ bias,    // (24,)
                   float*       __restrict__ out,     // (N, 32, 24)
                   int N) {
  // Per-wave LDS: first-square table (indexed by piece id 0..32) and the
  // gathered 32x10 feature tile (padded to 12 floats/row: bank-conflict-free
  // A-tile reads and K zero-padding for the 3rd K-step of the WMMA chain).
  __shared__ unsigned firstSq[WAVES_PER_BLOCK][36];
  __shared__ float    raw[WAVES_PER_BLOCK][P_DIM][12];

  const int lane = threadIdx.x & 31;
  const int wid  = threadIdx.x >> 5;
  const int half = lane >> 4;       // 0: lanes 0-15, 1: lanes 16-31
  const int lq   = lane & 15;

  unsigned* fs = firstSq[wid];
  float (*r)[12] = raw[wid];

  // ---- Build B (W^T in WMMA 4x16 layout) and bias-C once per wave ----
  // B row k lives in: VGPR0 = {K=0 lanes0-15, K=2 lanes16-31},
  //                   VGPR1 = {K=1, K=3}; N = lane%16.
  v2f  Breg[2][3];
  float biasv[2];
#pragma unroll
  for (int nt = 0; nt < 2; ++nt) {
    const int f = nt * 16 + lq;
    biasv[nt] = (f < F_DIM) ? bias[f] : 0.0f;
#pragma unroll
    for (int ks = 0; ks < 3; ++ks) {
      const int c0 = ks * 4 + half * 2;
      const float w0 = (f < F_DIM && c0     < C_DIM) ? Wm[f * C_DIM + c0]     : 0.0f;
      const float w1 = (f < F_DIM && c0 + 1 < C_DIM) ? Wm[f * C_DIM + c0 + 1] : 0.0f;
      Breg[nt][ks] = (v2f){w0, w1};
    }
  }

  const int wavesTotal = (int)gridDim.x * WAVES_PER_BLOCK;

  for (int n = (int)blockIdx.x * WAVES_PER_BLOCK + wid; n < N; n += wavesTotal) {
    // ---- 1) first-square-per-piece via LDS atomic min (ds_min_u32) ----
    // Same-wave LDS ops are in order: store -> atomic -> load needs no barrier.
    fs[lane] = 64u;
    if (lane == 0) fs[32] = 64u;
    const int s0 = 2 * lane, s1 = s0 + 1;
    const int id0 = ids[(size_t)n * SQ + s0];
    const int id1 = ids[(size_t)n * SQ + s1];
    atomicMin(&fs[id0], (unsigned)s0);
    atomicMin(&fs[id1], (unsigned)s1);

    // ---- 2) gather this lane's piece features (zero if absent) ----
    const unsigned sq = fs[lane + 1];           // piece id = lane+1
    const bool present = sq < 64u;
    const float* bp = board + (size_t)n * (C_DIM * SQ) + (present ? sq : 0u);
#pragma unroll
    for (int c = 0; c < C_DIM; ++c)
      r[lane][c] = present ? bp[c * SQ] : 0.0f;
    r[lane][10] = 0.0f;
    r[lane][11] = 0.0f;

    // ---- 3) out(32x24) = raw(32x10) @ W^T + b via f32 WMMA 16x16x4 ----
    float* outn = out + (size_t)n * (P_DIM * F_DIM);
#pragma unroll
    for (int mt = 0; mt < 2; ++mt) {
      const int mrow = mt * 16 + lq;            // A row this lane supplies
#pragma unroll
      for (int nt = 0; nt < 2; ++nt) {
        v8f acc = {biasv[nt], biasv[nt], biasv[nt], biasv[nt],
                   biasv[nt], biasv[nt], biasv[nt], biasv[nt]};
#pragma unroll
        for (int ks = 0; ks < 3; ++ks) {
          const int c0 = ks * 4 + half * 2;     // A: lanes0-15 K={4k,4k+1}, 16-31 K={4k+2,4k+3}
          const v2f a = (v2f){r[mrow][c0], r[mrow][c0 + 1]};
          acc = __builtin_amdgcn_wmma_f32_16x16x4_f32(
              /*neg_a=*/false, a, /*neg_b=*/false, Breg[nt][ks],
              /*c_mod=*/(short)0, acc, /*reuse_a=*/false, /*reuse_b=*/false);
        }
        // ---- 4) store D: lane -> f, vgpr v -> m = 16*mt + 8*half + v ----
        const int f = nt * 16 + lq;
        if (f < F_DIM) {
          const int mbase = mt * 16 + half * 8;
#pragma unroll
          for (int v = 0; v < 8; ++v)
            outn[(mbase + v) * F_DIM + f] = acc[v];
        }
      }
    }
  }
}

extern "C" void kernel_launch(void* const* d_in, const int* in_sizes, int n_in,
                              void* d_out, int out_size, void* d_ws, size_t ws_size,
                              hipStream_t stream) {
  const float* board = (const float*)d_in[0];   // (N,10,8,8) f32
  const int*   ids   = (const int*)d_in[1];     // (N,8,8) int
  const float* Wm    = (const float*)d_in[2];   // (24,10) f32
  const float* bias  = (const float*)d_in[3];   // (24,) f32
  float* out = (float*)d_out;                   // (N,32,24) f32

  const int N = in_sizes[0] / (C_DIM * SQ);     // 131072
  int blocks = (N + WAVES_PER_BLOCK - 1) / WAVES_PER_BLOCK;
  if (blocks > 2048) blocks = 2048;             // grid-stride: ~8 boards/wave,
                                                // amortizes B/bias setup
  piece_extract_wmma<<<blocks, 256, 0, stream>>>(board, ids, Wm, bias, out, N);
}